// Decoder_31009663877186
// MI455X (gfx1250) — compile-verified
//
#include <hip/hip_runtime.h>

typedef unsigned long long u64;

#define B_    16
#define C_    80
#define H_    128
#define W_    128
#define HW_   (H_ * W_)
#define K_    100
#define CK_   (C_ * K_)     // 8000 candidates per batch in stage 2
#define TPB   256           // 8 wave32 waves per workgroup
#define EPT   (HW_ / TPB)   // 64 heat elements owned per thread

// ---- probe for the CDNA5 async global->LDS path (ASYNCcnt) ----
#if defined(__has_builtin)
#if __has_builtin(__builtin_amdgcn_global_load_async_to_lds_b128) && \
    __has_builtin(__builtin_amdgcn_s_wait_asynccnt)
#define USE_ASYNC_LDS 1
#endif
#endif
#ifndef USE_ASYNC_LDS
#define USE_ASYNC_LDS 0
#endif

#if USE_ASYNC_LDS
// Pointee type from the round-1 diagnostic:
//   "__attribute__((__vector_size__(4 * sizeof(int)))) int __device__ *"
// i.e. 16-byte int vector in AS1 (global) / AS3 (LDS).
typedef int v4i __attribute__((vector_size(4 * sizeof(int))));
typedef __attribute__((address_space(1))) v4i gv4i;
typedef __attribute__((address_space(3))) v4i lv4i;
#endif

// Order-preserving key: monotone float transform in high 32 bits, ~index in
// low 32 bits so equal values tie-break toward the LOWER index (lax.top_k
// stability). All live values are >= 0; extracted slots are set to -1.0f,
// which maps below +0.0 under the transform.
__device__ __forceinline__ u64 make_key(float v, int idx) {
    unsigned b  = __float_as_uint(v);
    unsigned ob = (b & 0x80000000u) ? ~b : (b | 0x80000000u);
    return ((u64)ob << 32) | (u64)(unsigned)(~(unsigned)idx);
}

// Block-wide max over 256 per-thread keys via LDS tree reduction.
__device__ __forceinline__ u64 reduce_max_key(u64* red, u64 v, int tid) {
    red[tid] = v;
    __syncthreads();
#pragma unroll
    for (int s = TPB / 2; s >= 1; s >>= 1) {
        if (tid < s) {
            u64 o = red[tid + s];
            if (o > red[tid]) red[tid] = o;
        }
        __syncthreads();
    }
    return red[0];
}

// ---------------------------------------------------------------------------
// Kernel 1: per-(b,c) NMS + top-100 over 128x128 map, entirely out of LDS.
// LDS: 64KB raw heat + 64KB NMS values + 4KB selection scratch (~132KB,
// 2 workgroups per 320KB WGP on CDNA5).
// ---------------------------------------------------------------------------
extern "C" __global__ void __launch_bounds__(TPB)
nms_topk_class_kernel(const float* __restrict__ heat,
                      float* __restrict__ scores_c,   // [B, C*K]
                      int*   __restrict__ inds_c)     // [B, C*K]
{
    extern __shared__ unsigned char smem[];
    float* hs  = (float*)smem;          // HW_ floats: raw heat map
    float* ns  = hs + HW_;              // HW_ floats: NMS-suppressed values
    u64*   pk  = (u64*)(ns + HW_);      // TPB per-thread partial-max keys
    u64*   red = pk + TPB;              // TPB reduction scratch

    const int tid = threadIdx.x;
    const int bc  = blockIdx.x;         // b*C + c
    const float* __restrict__ src = heat + (size_t)bc * HW_;

    // ---- stage the 64KB map into LDS (single HBM touch per element) ----
    // Lane layout: lane base = tid*16B; 16 chunks, stride 256 lanes * 16B = 4KB.
#if USE_ASYNC_LDS
    {
        gv4i* g = (gv4i*)((unsigned long long)(src) + (unsigned long long)tid * 16u);
        lv4i* l = (lv4i*)((unsigned)(unsigned long long)(void*)hs + (unsigned)tid * 16u);
        // 16 async transfers with immediate IOFFSETs (no loop versioning,
        // clause-friendly back-to-back issue); single ASYNCcnt wait at the end.
#define ASYNC_CHUNK(n) __builtin_amdgcn_global_load_async_to_lds_b128(g, l, (n) * 4096, 0)
        ASYNC_CHUNK(0);  ASYNC_CHUNK(1);  ASYNC_CHUNK(2);  ASYNC_CHUNK(3);
        ASYNC_CHUNK(4);  ASYNC_CHUNK(5);  ASYNC_CHUNK(6);  ASYNC_CHUNK(7);
        ASYNC_CHUNK(8);  ASYNC_CHUNK(9);  ASYNC_CHUNK(10); ASYNC_CHUNK(11);
        ASYNC_CHUNK(12); ASYNC_CHUNK(13); ASYNC_CHUNK(14); ASYNC_CHUNK(15);
#undef ASYNC_CHUNK
        __builtin_amdgcn_s_wait_asynccnt(0);
    }
#else
    {
        const float4* s4 = (const float4*)src;
        float4*       h4 = (float4*)hs;
        for (int i = tid; i < HW_ / 4; i += TPB) h4[i] = s4[i];
    }
#endif
    __syncthreads();

    // ---- NMS: v = (3x3 max == h) ? h : 0  (out-of-bounds = -inf) ----
    for (int i = 0; i < EPT; ++i) {
        int   idx = tid * EPT + i;
        int   y = idx >> 7, x = idx & (W_ - 1);
        float h = hs[idx];
        float m = h;
        int y0 = y > 0 ? y - 1 : 0, y1 = y < H_ - 1 ? y + 1 : H_ - 1;
        int x0 = x > 0 ? x - 1 : 0, x1 = x < W_ - 1 ? x + 1 : W_ - 1;
        for (int yy = y0; yy <= y1; ++yy)
            for (int xx = x0; xx <= x1; ++xx) {
                float v = hs[(yy << 7) + xx];
                m = v > m ? v : m;
            }
        ns[idx] = (m == h) ? h : 0.0f;
    }
    __syncthreads();

    // ---- per-thread partial max over owned 64 elements ----
    u64 best = 0;
    for (int i = 0; i < EPT; ++i) {
        int idx = tid * EPT + i;
        u64 kk  = make_key(ns[idx], idx);
        if (kk > best) best = kk;
    }
    pk[tid] = best;
    __syncthreads();

    // ---- iterative top-100 extraction ----
    for (int k = 0; k < K_; ++k) {
        u64 win = reduce_max_key(red, pk[tid], tid);
        int idx = (int)(~(unsigned)win);
        if (tid == 0) {
            unsigned ob = (unsigned)(win >> 32);
            scores_c[(size_t)bc * K_ + k] = __uint_as_float(ob & 0x7FFFFFFFu);
            inds_c[(size_t)bc * K_ + k]   = idx;
        }
        if ((idx >> 6) == tid) {            // EPT == 64: owner rescans
            ns[idx] = -1.0f;
            u64 nb = 0;
            for (int i = 0; i < EPT; ++i) {
                int j  = tid * EPT + i;
                u64 kk = make_key(ns[j], j);
                if (kk > nb) nb = kk;
            }
            pk[tid] = nb;
        }
        __syncthreads();
    }
}

// ---------------------------------------------------------------------------
// Kernel 2: per-batch top-100 over 8000 (class,k) candidates + gather + bbox.
// ---------------------------------------------------------------------------
extern "C" __global__ void __launch_bounds__(TPB)
topk_all_kernel(const float* __restrict__ scores_c,  // [B, C*K]
                const int*   __restrict__ inds_c,    // [B, C*K]
                const float* __restrict__ wh,        // [B, 2, H, W]
                const float* __restrict__ reg,       // [B, 2, H, W]
                float* __restrict__ out)             // [B, K, 6]
{
    extern __shared__ unsigned char smem[];
    float* ss  = (float*)smem;          // CK_ floats
    u64*   pk  = (u64*)(ss + CK_);      // TPB
    u64*   red = pk + TPB;              // TPB

    const int tid = threadIdx.x;
    const int b   = blockIdx.x;
    const float* __restrict__ sb = scores_c + (size_t)b * CK_;

    for (int i = tid; i < CK_; i += TPB) ss[i] = sb[i];
    __syncthreads();

    // strided ownership: thread t owns candidates {t, t+256, ...}
    u64 best = 0;
    for (int f = tid; f < CK_; f += TPB) {
        u64 kk = make_key(ss[f], f);
        if (kk > best) best = kk;
    }
    pk[tid] = best;
    __syncthreads();

    for (int k = 0; k < K_; ++k) {
        u64 win = reduce_max_key(red, pk[tid], tid);
        int f   = (int)(~(unsigned)win);
        if (tid == 0) {
            float score = __uint_as_float(((unsigned)(win >> 32)) & 0x7FFFFFFFu);
            int   cls   = f / K_;                         // topk_ind // K
            int   ind   = inds_c[(size_t)b * CK_ + f];    // spatial index
            float y = (float)(ind >> 7);
            float x = (float)(ind & (W_ - 1));
            // reg_flat[b, ind, ch] == reg[b, ch, ind] (NCHW), same for wh
            size_t p0 = ((size_t)b * 2) * HW_ + (size_t)ind;
            size_t p1 = p0 + HW_;
            float rx = reg[p0], ry = reg[p1];
            float ww = wh[p0],  hh = wh[p1];
            float xs = x + rx,  ys = y + ry;
            float* o = out + ((size_t)b * K_ + k) * 6;
            o[0] = xs - ww * 0.5f;
            o[1] = ys - hh * 0.5f;
            o[2] = xs + ww * 0.5f;
            o[3] = ys + hh * 0.5f;
            o[4] = score;
            o[5] = (float)cls;
        }
        if ((f % TPB) == tid) {             // owner invalidates + rescans
            ss[f] = -1.0f;
            u64 nb = 0;
            for (int j = tid; j < CK_; j += TPB) {
                u64 kk = make_key(ss[j], j);
                if (kk > nb) nb = kk;
            }
            pk[tid] = nb;
        }
        __syncthreads();
    }
}

// ---------------------------------------------------------------------------
extern "C" void kernel_launch(void* const* d_in, const int* in_sizes, int n_in,
                              void* d_out, int out_size, void* d_ws, size_t ws_size,
                              hipStream_t stream) {
    (void)in_sizes; (void)n_in; (void)out_size; (void)ws_size;
    const float* heat = (const float*)d_in[0];
    const float* wh   = (const float*)d_in[1];
    const float* reg  = (const float*)d_in[2];
    float*       out  = (float*)d_out;

    // workspace: scores_c [B*C*K] floats (512KB) then inds_c [B*C*K] ints (512KB)
    float* scores_c = (float*)d_ws;
    int*   inds_c   = (int*)((char*)d_ws + (size_t)B_ * CK_ * sizeof(float));

    const size_t smem1 = (size_t)HW_ * 2 * sizeof(float) + 2 * TPB * sizeof(u64); // ~132KB
    const size_t smem2 = (size_t)CK_ * sizeof(float)     + 2 * TPB * sizeof(u64); // ~36KB

    // raise dynamic-LDS cap (>64KB is legal on CDNA5's 320KB WGP LDS);
    // host-side attribute set, deterministic and graph-capture safe.
    (void)hipFuncSetAttribute((const void*)nms_topk_class_kernel,
                              hipFuncAttributeMaxDynamicSharedMemorySize, (int)smem1);

    nms_topk_class_kernel<<<B_ * C_, TPB, smem1, stream>>>(heat, scores_c, inds_c);
    topk_all_kernel<<<B_, TPB, smem2, stream>>>(scores_c, inds_c, wh, reg, out);
}